// GaussianRasterizer_14525579395201
// MI455X (gfx1250) — compile-verified
//
#include <hip/hip_runtime.h>

// ---------------- problem constants (match reference) ----------------
#define NGAUSS   2048
#define IMG_H    160
#define IMG_W    224
#define IMG_HW   (IMG_H * IMG_W)          // 35840
#define TANFOVX  0.5774f
#define TANFOVY  (0.5774f * 160.0f / 224.0f)
#define NEARCLIP 0.2f
#define LOWPASS  0.3f
#define ALPHA_MIN (1.0f / 255.0f)
#define CHNK     256                      // gaussians staged in LDS per pass
#define NCHUNK   (NGAUSS / CHNK)          // 8

// output layout (flat, in return order): color(3*HW) radii(N,int) depth(HW) opacity(HW) n_touched(N,int)
#define OFF_RADII  (3 * IMG_HW)
#define OFF_DEPTH  (3 * IMG_HW + NGAUSS)
#define OFF_OPAC   (3 * IMG_HW + NGAUSS + IMG_HW)
#define OFF_NTOUCH (3 * IMG_HW + NGAUSS + 2 * IMG_HW)

typedef __attribute__((ext_vector_type(2))) float v2f;
typedef __attribute__((ext_vector_type(8))) float v8f;

// ---------------- kernel 1: per-gaussian projection / conic ----------------
__global__ void gs_preprocess(const float* __restrict__ means3D,
                              const float* __restrict__ opac,
                              const float* __restrict__ colors,
                              const float* __restrict__ scales,
                              const float* __restrict__ rots,
                              const float* __restrict__ vm,
                              float4* __restrict__ preA,   // px,py,con0,con1
                              float4* __restrict__ preB,   // con2,op,valid,orig_idx(bits)
                              float4* __restrict__ preC,   // r,g,b,depth
                              float* __restrict__ key,
                              int* __restrict__ radii) {
  int i = blockIdx.x * blockDim.x + threadIdx.x;
  if (i >= NGAUSS) return;

  float m0 = means3D[3 * i], m1 = means3D[3 * i + 1], m2 = means3D[3 * i + 2];
  // t = Rw @ mean + tw  (viewmatrix row-major 4x4)
  float t0 = vm[0] * m0 + vm[1] * m1 + vm[2]  * m2 + vm[3];
  float t1 = vm[4] * m0 + vm[5] * m1 + vm[6]  * m2 + vm[7];
  float t2 = vm[8] * m0 + vm[9] * m1 + vm[10] * m2 + vm[11];

  const float fx = IMG_W / (2.0f * TANFOVX);
  const float fy = IMG_H / (2.0f * TANFOVY);
  const float limx = 1.3f * TANFOVX, limy = 1.3f * TANFOVY;
  float txz = fminf(fmaxf(t0 / t2, -limx), limx) * t2;
  float tyz = fminf(fmaxf(t1 / t2, -limy), limy) * t2;

  // quaternion -> rotation
  float qw = rots[4 * i], qx = rots[4 * i + 1], qy = rots[4 * i + 2], qz = rots[4 * i + 3];
  float qn = rsqrtf(qw * qw + qx * qx + qy * qy + qz * qz);
  qw *= qn; qx *= qn; qy *= qn; qz *= qn;
  float R00 = 1.f - 2.f * (qy * qy + qz * qz), R01 = 2.f * (qx * qy - qw * qz), R02 = 2.f * (qx * qz + qw * qy);
  float R10 = 2.f * (qx * qy + qw * qz), R11 = 1.f - 2.f * (qx * qx + qz * qz), R12 = 2.f * (qy * qz - qw * qx);
  float R20 = 2.f * (qx * qz - qw * qy), R21 = 2.f * (qy * qz + qw * qx), R22 = 1.f - 2.f * (qx * qx + qy * qy);

  float s0 = scales[3 * i], s1 = scales[3 * i + 1], s2 = scales[3 * i + 2];
  float S0 = s0 * s0, S1 = s1 * s1, S2 = s2 * s2;   // SCALE_MOD = 1

  // Sigma = R diag(S) R^T (symmetric 3x3)
  float Sg00 = R00 * R00 * S0 + R01 * R01 * S1 + R02 * R02 * S2;
  float Sg01 = R00 * R10 * S0 + R01 * R11 * S1 + R02 * R12 * S2;
  float Sg02 = R00 * R20 * S0 + R01 * R21 * S1 + R02 * R22 * S2;
  float Sg11 = R10 * R10 * S0 + R11 * R11 * S1 + R12 * R12 * S2;
  float Sg12 = R10 * R20 * S0 + R11 * R21 * S1 + R12 * R22 * S2;
  float Sg22 = R20 * R20 * S0 + R21 * R21 * S1 + R22 * R22 * S2;

  float iz = 1.0f / t2, iz2 = iz * iz;
  float J00 = fx * iz, J02 = -fx * txz * iz2;
  float J11 = fy * iz, J12 = -fy * tyz * iz2;

  // Tm = J @ Rw
  float T00 = J00 * vm[0] + J02 * vm[8];
  float T01 = J00 * vm[1] + J02 * vm[9];
  float T02 = J00 * vm[2] + J02 * vm[10];
  float T10 = J11 * vm[4] + J12 * vm[8];
  float T11 = J11 * vm[5] + J12 * vm[9];
  float T12 = J11 * vm[6] + J12 * vm[10];

  // cov2 = Tm Sigma Tm^T
  float U00 = T00 * Sg00 + T01 * Sg01 + T02 * Sg02;
  float U01 = T00 * Sg01 + T01 * Sg11 + T02 * Sg12;
  float U02 = T00 * Sg02 + T01 * Sg12 + T02 * Sg22;
  float U10 = T10 * Sg00 + T11 * Sg01 + T12 * Sg02;
  float U11 = T10 * Sg01 + T11 * Sg11 + T12 * Sg12;
  float U12 = T10 * Sg02 + T11 * Sg12 + T12 * Sg22;
  float a = U00 * T00 + U01 * T01 + U02 * T02 + LOWPASS;
  float b = U00 * T10 + U01 * T11 + U02 * T12;
  float c = U10 * T10 + U11 * T11 + U12 * T12 + LOWPASS;

  float det = a * c - b * b;
  float inv = 1.0f / det;
  float con0 = c * inv, con1 = -b * inv, con2 = a * inv;
  float mid = 0.5f * (a + c);
  float lam1 = mid + sqrtf(fmaxf(0.1f, mid * mid - det));
  float radf = ceilf(3.0f * sqrtf(lam1));

  float px = ((t0 / (t2 * TANFOVX) + 1.0f) * IMG_W - 1.0f) * 0.5f;
  float py = ((t1 / (t2 * TANFOVY) + 1.0f) * IMG_H - 1.0f) * 0.5f;

  bool valid = (t2 > NEARCLIP) && (det > 0.0f) && (radf > 0.0f);
  radii[i] = valid ? (int)radf : 0;
  key[i] = valid ? t2 : __builtin_inff();
  preA[i] = make_float4(px, py, con0, con1);
  preB[i] = make_float4(con2, opac[i], valid ? 1.0f : 0.0f, __int_as_float(i));
  preC[i] = make_float4(colors[3 * i], colors[3 * i + 1], colors[3 * i + 2], t2);
}

// ---------------- kernel 2: stable O(N^2) rank sort by depth, scatter ----------------
// colT is written channel-major per chunk: colT[chunk][ch][pos], so the raster
// kernel can async-copy it byte-for-byte into LDS.
__global__ void gs_sort_scatter(const float* __restrict__ key,
                                const float4* __restrict__ preA,
                                const float4* __restrict__ preB,
                                const float4* __restrict__ preC,
                                float4* __restrict__ sA,
                                float4* __restrict__ sB,
                                float* __restrict__ colT) {
  __shared__ float sk[NGAUSS];
  for (int j = threadIdx.x; j < NGAUSS; j += blockDim.x) sk[j] = key[j];
  __syncthreads();
  int i = blockIdx.x * blockDim.x + threadIdx.x;
  if (i >= NGAUSS) return;
  float ki = sk[i];
  int r = 0;
  for (int j = 0; j < NGAUSS; ++j) {
    float kj = sk[j];
    r += (kj < ki) || ((kj == ki) && (j < i));   // stable rank
  }
  sA[r] = preA[i];
  sB[r] = preB[i];
  float4 pc = preC[i];
  int chunk = r >> 8;            // CHNK = 256
  int pos = r & (CHNK - 1);
  float* ct = colT + chunk * (4 * CHNK);
  ct[0 * CHNK + pos] = pc.x;
  ct[1 * CHNK + pos] = pc.y;
  ct[2 * CHNK + pos] = pc.z;
  ct[3 * CHNK + pos] = pc.w;
}

// ---------------- kernel 3: zero n_touched ----------------
__global__ void gs_zero_ntouch(int* __restrict__ nt) {
  int i = blockIdx.x * blockDim.x + threadIdx.x;
  if (i < NGAUSS) nt[i] = 0;
}

// ---------------- async global->LDS copy (CDNA5, ASYNCcnt-tracked) ----------------
__device__ __forceinline__ void async_b128(void* lds, const void* g) {
  // LDS byte address = low 32 bits of the flat shared pointer (ISA 10.2 aperture rule)
  unsigned l = (unsigned)(unsigned long long)lds;
  // GV mode: 64-bit global address in VGPR pair, saddr = off
  asm volatile("global_load_async_to_lds_b128 %0, %1, off"
               :: "v"(l), "v"(g) : "memory");
}

// ---------------- kernel 4: rasterize (WMMA f32 16x16x4 accumulation) ----------------
__device__ __forceinline__ float gs_alpha(float4 ga, float4 gb, float gx, float gy) {
  float dx = ga.x - gx, dy = ga.y - gy;
  float power = -0.5f * (ga.z * dx * dx + gb.x * dy * dy) - ga.w * dx * dy;
  float alpha = fminf(0.99f, gb.y * __expf(power));
  bool ok = (power <= 0.0f) && (alpha >= ALPHA_MIN) && (gb.z > 0.5f);
  return ok ? alpha : 0.0f;
}

__global__ void __launch_bounds__(256)
gs_raster(const float4* __restrict__ sA,
          const float4* __restrict__ sB,
          const float* __restrict__ colT,  // channel-major per 256-gaussian chunk
          const float* __restrict__ bg,
          float* __restrict__ out_color,   // 3*HW
          float* __restrict__ out_depth,   // HW
          float* __restrict__ out_opac,    // HW
          int* __restrict__ out_ntouch) {  // N
  // double-buffered LDS tiles (24 KB total)
  __shared__ float4 shA[2][CHNK];
  __shared__ float4 shB[2][CHNK];
  __shared__ float  shC[2][4 * CHNK];      // channel-major: [ch*CHNK + g] = {r,g,b,depth}

  const int tid  = threadIdx.x;
  const int lane = tid & 31;
  const int wv   = tid >> 5;               // 8 waves per block
  const int pixBase = blockIdx.x * 128 + wv * 16;   // 16 pixels per wave
  const int p = pixBase + (lane & 15);     // lanes 0-15 and 16-31 mirror the same pixels
  const float gxf = (float)(p % IMG_W);
  const float gyf = (float)(p / IMG_W);
  const bool hi = (lane >= 16);
  const int  m  = lane & 15;               // A-matrix row this lane feeds (channel)

  float T = 1.0f;                          // transmittance
  v8f acc = {};                            // rows 0..3 = R,G,B,depth sums for 16 pixels

  // issue one chunk's staging: 3 async B128 per thread (geoA, geoB, colT = 12 KB/chunk)
  auto issue_chunk = [&](int cb, int buf) {
    async_b128(&shA[buf][tid], sA + cb * CHNK + tid);
    async_b128(&shB[buf][tid], sB + cb * CHNK + tid);
    async_b128(&shC[buf][tid * 4], colT + cb * (4 * CHNK) + tid * 4);
  };

  issue_chunk(0, 0);                       // prologue prefetch

  for (int cb = 0; cb < NCHUNK; ++cb) {
    const int cur = cb & 1;
    if (cb + 1 < NCHUNK) {
      issue_chunk(cb + 1, 1 - cur);        // prefetch next chunk into other buffer
      // 3 outstanding = the prefetch just issued; current chunk's 3 are complete
      asm volatile("s_wait_asynccnt 0x3" ::: "memory");
    } else {
      asm volatile("s_wait_asynccnt 0x0" ::: "memory");
    }
    __syncthreads();                       // all waves' async data visible in LDS

    const float4* A = shA[cur];
    const float4* B = shB[cur];
    const float*  C = shC[cur];

    for (int gg = 0; gg < CHNK; gg += 4) {
      float4 A0 = A[gg + 0], A1 = A[gg + 1], A2 = A[gg + 2], A3 = A[gg + 3];
      float4 B0 = B[gg + 0], B1 = B[gg + 1], B2 = B[gg + 2], B3 = B[gg + 3];

      float a0 = gs_alpha(A0, B0, gxf, gyf);
      float a1 = gs_alpha(A1, B1, gxf, gyf);
      float a2 = gs_alpha(A2, B2, gxf, gyf);
      float a3 = gs_alpha(A3, B3, gxf, gyf);

      // n_touched: count lanes 0-15 only (16-31 are mirrors)
      unsigned long long bl0 = __ballot(a0 > 0.0f);
      unsigned long long bl1 = __ballot(a1 > 0.0f);
      unsigned long long bl2 = __ballot(a2 > 0.0f);
      unsigned long long bl3 = __ballot(a3 > 0.0f);
      if (lane == 0) {
        unsigned mm;
        mm = (unsigned)bl0 & 0xFFFFu; if (mm) atomicAdd(&out_ntouch[__float_as_int(B0.w)], (int)__popc(mm));
        mm = (unsigned)bl1 & 0xFFFFu; if (mm) atomicAdd(&out_ntouch[__float_as_int(B1.w)], (int)__popc(mm));
        mm = (unsigned)bl2 & 0xFFFFu; if (mm) atomicAdd(&out_ntouch[__float_as_int(B2.w)], (int)__popc(mm));
        mm = (unsigned)bl3 & 0xFFFFu; if (mm) atomicAdd(&out_ntouch[__float_as_int(B3.w)], (int)__popc(mm));
      }

      // sequential transmittance recurrence (front-to-back)
      float w0 = a0 * T; T *= 1.0f - a0;
      float w1 = a1 * T; T *= 1.0f - a1;
      float w2 = a2 * T; T *= 1.0f - a2;
      float w3 = a3 * T; T *= 1.0f - a3;

      // B matrix (4x16 f32): VGPR0 = K0 (lanes 0-15) / K2 (16-31), VGPR1 = K1 / K3
      v2f Bv;
      Bv.x = hi ? w2 : w0;
      Bv.y = hi ? w3 : w1;

      // A matrix (16x4 f32): lane m<4 supplies channel m of gaussians (K0,K1) or (K2,K3)
      v2f Av; Av.x = 0.0f; Av.y = 0.0f;
      if (m < 4) {
        int gsel = gg + (hi ? 2 : 0);
        Av.x = C[m * CHNK + gsel];
        Av.y = C[m * CHNK + gsel + 1];
      }

      // D = A x B + C  on the matrix pipe, full fp32
      acc = __builtin_amdgcn_wmma_f32_16x16x4_f32(
          /*neg_a=*/false, Av, /*neg_b=*/false, Bv,
          /*c_mod=*/(short)0, acc, /*reuse_a=*/false, /*reuse_b=*/false);
    }
    __syncthreads();   // all waves done reading buf `cur` before it is refilled
  }

  // D layout: lanes 0-15 VGPR j = row j (channels 0..3 used), pixel = lane
  if (lane < 16) {
    float b0 = bg[0], b1 = bg[1], b2 = bg[2];
    out_color[0 * IMG_HW + p] = acc[0] + b0 * T;
    out_color[1 * IMG_HW + p] = acc[1] + b1 * T;
    out_color[2 * IMG_HW + p] = acc[2] + b2 * T;
    out_depth[p] = acc[3];
    out_opac[p]  = 1.0f - T;
  }
}

// ---------------- host-side launch ----------------
extern "C" void kernel_launch(void* const* d_in, const int* in_sizes, int n_in,
                              void* d_out, int out_size, void* d_ws, size_t ws_size,
                              hipStream_t stream) {
  const float* means3D = (const float*)d_in[0];
  // d_in[1] = means2D (unused placeholder in reference)
  const float* opac    = (const float*)d_in[2];
  const float* colors  = (const float*)d_in[3];
  const float* scales  = (const float*)d_in[4];
  const float* rots    = (const float*)d_in[5];
  const float* bg      = (const float*)d_in[6];
  const float* vm      = (const float*)d_in[7];

  // workspace layout
  float4* preA = (float4*)d_ws;
  float4* preB = preA + NGAUSS;
  float4* preC = preB + NGAUSS;
  float4* sA   = preC + NGAUSS;
  float4* sB   = sA + NGAUSS;
  float*  colT = (float*)(sB + NGAUSS);    // 4*NGAUSS floats, channel-major per chunk
  float*  key  = colT + 4 * NGAUSS;

  float* out     = (float*)d_out;
  int*   radii   = (int*)d_out + OFF_RADII;
  float* depth   = out + OFF_DEPTH;
  float* opacm   = out + OFF_OPAC;
  int*   ntouch  = (int*)d_out + OFF_NTOUCH;

  gs_preprocess<<<NGAUSS / 256, 256, 0, stream>>>(means3D, opac, colors, scales, rots, vm,
                                                  preA, preB, preC, key, radii);
  gs_sort_scatter<<<NGAUSS / 256, 256, 0, stream>>>(key, preA, preB, preC, sA, sB, colT);
  gs_zero_ntouch<<<NGAUSS / 256, 256, 0, stream>>>(ntouch);
  gs_raster<<<IMG_HW / 128, 256, 0, stream>>>(sA, sB, colT, bg, out, depth, opacm, ntouch);
}